// GAT_41910290874823
// MI455X (gfx1250) — compile-verified
//
#include <hip/hip_runtime.h>
#include <cfloat>
#include <cmath>

typedef __attribute__((ext_vector_type(2))) float v2f;
typedef __attribute__((ext_vector_type(8))) float v8f;

// ---------------------------------------------------------------------------
// float atomic max via signed/unsigned int atomics (standard IEEE trick):
// non-negative values compare correctly as signed ints; negative values
// compare in reverse as unsigned ints.
// ---------------------------------------------------------------------------
__device__ inline void atomicMaxFloat(float* addr, float val) {
  if (val >= 0.0f) {
    atomicMax((int*)addr, __float_as_int(val));
  } else {
    atomicMin((unsigned int*)addr, (unsigned int)__float_as_int(val));
  }
}

// ---------------------------------------------------------------------------
// One 16-row GEMM tile: Y = X @ W^T + bias via V_WMMA_F32_16X16X4_F32.
// FULL=true means all 16 rows are in range (uniform wave-level fact):
// no row clamps/selects and unguarded stores.
// A frag (16x4 f32): lane m = lane&15, K = k0 + 2*(lane>=16) + {0,1}
// B frag (4x16 f32): lane n = lane&15, K = k0 + 2*(lane>=16) + {0,1}
// C/D (16x16 f32):   VGPR r -> M = r + 8*(lane>=16), N = lane&15
// ---------------------------------------------------------------------------
template <int DIN, int DOUT, bool FULL>
__device__ __forceinline__ void gemm_tile(const float* __restrict__ X,
                                          const float* __restrict__ W,
                                          const float* __restrict__ bias,
                                          float* __restrict__ Y,
                                          int nrows, int row0, int lane) {
  constexpr int NT = (DOUT + 15) / 16;  // output 16-col tiles
  const int g  = lane >> 4;             // lane group
  const int ln = lane & 15;

  v8f acc[NT] = {};

  const int  m  = row0 + ln;
  const bool mv = FULL ? true : (m < nrows);
  const int  mc = FULL ? m : (mv ? m : (nrows - 1));
  const float* Xrow = X + (size_t)mc * DIN;

#pragma unroll
  for (int k0 = 0; k0 < DIN; k0 += 4) {
    const int  ka  = k0 + 2 * g;
    const bool k0v = (ka < DIN);
    const bool k1v = (ka + 1 < DIN);
    const int  ka0 = k0v ? ka : (DIN - 1);   // clamped k indices
    const int  ka1 = k1v ? (ka + 1) : (DIN - 1);
    const float ax = Xrow[ka0];              // unconditional loads
    const float ay = Xrow[ka1];
    v2f a;
    a.x = (mv && k0v) ? ax : 0.0f;           // v_cndmask, no branch
    a.y = (mv && k1v) ? ay : 0.0f;
#pragma unroll
    for (int t = 0; t < NT; ++t) {
      const int  d  = t * 16 + ln;
      const bool dv = (d < DOUT);
      const int  dc = dv ? d : (DOUT - 1);
      const float* Wrow = W + (size_t)dc * DIN;
      const float bx = Wrow[ka0];
      const float by = Wrow[ka1];
      v2f bb;
      bb.x = (dv && k0v) ? bx : 0.0f;
      bb.y = (dv && k1v) ? by : 0.0f;
      acc[t] = __builtin_amdgcn_wmma_f32_16x16x4_f32(
          /*neg_a=*/false, a, /*neg_b=*/false, bb,
          /*c_mod=*/(short)0, acc[t], /*reuse_a=*/false, /*reuse_b=*/false);
    }
  }

#pragma unroll
  for (int t = 0; t < NT; ++t) {
    const int d = t * 16 + ln;
    if (d >= DOUT) continue;                 // compile-time for DOUT%16==0
    const float bv = bias[d];
#pragma unroll
    for (int r = 0; r < 8; ++r) {
      const int row = row0 + r + 8 * g;
      if (FULL || row < nrows) Y[(size_t)row * DOUT + d] = acc[t][r] + bv;
    }
  }
}

template <int DIN, int DOUT>
__global__ void gemm_bias_wmma(const float* __restrict__ X,
                               const float* __restrict__ W,
                               const float* __restrict__ bias,
                               float* __restrict__ Y, int nrows) {
  const int wave = (int)((blockIdx.x * blockDim.x + threadIdx.x) >> 5);
  const int lane = (int)(threadIdx.x & 31);
  const int row0 = wave * 16;
  if (row0 >= nrows) return;            // uniform per wave
  if (row0 + 16 <= nrows) {             // uniform per wave: fast path
    gemm_tile<DIN, DOUT, true>(X, W, bias, Y, nrows, row0, lane);
  } else {
    gemm_tile<DIN, DOUT, false>(X, W, bias, Y, nrows, row0, lane);
  }
}

// ---------------------------------------------------------------------------
// Edge pass 1: logits[e][h] = att[h] . leaky_relu(xl[src]+xr[dst], 0.2)
// and segment max over dst via float atomicMax. Rows loaded as float4
// (H*C is a multiple of 4; all workspace rows are 16B aligned).
// ---------------------------------------------------------------------------
template <int H, int C>
__global__ void edge_logits_max(const float* __restrict__ xl,
                                const float* __restrict__ xr,
                                const int* __restrict__ src,
                                const int* __restrict__ dst,
                                const float* __restrict__ att,
                                float* __restrict__ logits,
                                float* __restrict__ mx, int nE) {
  constexpr int HC = H * C;
  static_assert(HC % 4 == 0, "row must be float4-size");
  const int e = blockIdx.x * blockDim.x + threadIdx.x;
  if (e >= nE) return;
  const int s = src[e], d = dst[e];
  const float4* pl = (const float4*)(xl + (size_t)s * HC);
  const float4* pr = (const float4*)(xr + (size_t)d * HC);
  float v[HC];
#pragma unroll
  for (int i = 0; i < HC / 4; ++i) {
    const float4 a = pl[i];
    const float4 b = pr[i];
    v[4 * i + 0] = a.x + b.x;
    v[4 * i + 1] = a.y + b.y;
    v[4 * i + 2] = a.z + b.z;
    v[4 * i + 3] = a.w + b.w;
  }
#pragma unroll
  for (int h = 0; h < H; ++h) {
    float acc = 0.0f;
#pragma unroll
    for (int c = 0; c < C; ++c) {
      float t = v[h * C + c];
      t = (t > 0.0f) ? t : 0.2f * t;  // leaky_relu slope 0.2
      acc += att[h * C + c] * t;
    }
    logits[(size_t)e * H + h] = acc;
    atomicMaxFloat(&mx[(size_t)d * H + h], acc);
  }
}

// ---------------------------------------------------------------------------
// Edge pass 2: p = exp(logit - m[dst]); denom[dst] += p  (in-place in logits)
// ---------------------------------------------------------------------------
template <int H>
__global__ void edge_exp_denom(float* __restrict__ logits,
                               const float* __restrict__ mx,
                               const int* __restrict__ dst,
                               float* __restrict__ dn, int nEH) {
  const int t = blockIdx.x * blockDim.x + threadIdx.x;
  if (t >= nEH) return;
  const int e = t / H;      // H is a compile-time power of two -> shift
  const int h = t - e * H;
  const int d = dst[e];
  const float p = expf(logits[t] - mx[(size_t)d * H + h]);
  logits[t] = p;
  atomicAdd(&dn[(size_t)d * H + h], p);
}

// ---------------------------------------------------------------------------
// Edge pass 3: acc[dst] += (p/denom[dst]) * xl[src]   (per edge,head)
// ---------------------------------------------------------------------------
template <int H, int C>
__global__ void edge_aggregate(const float* __restrict__ p,
                               const float* __restrict__ dn,
                               const float* __restrict__ xl,
                               const int* __restrict__ src,
                               const int* __restrict__ dst,
                               float* __restrict__ acc, int nE) {
  const int t = blockIdx.x * blockDim.x + threadIdx.x;
  if (t >= nE * H) return;
  const int e = t / H;
  const int h = t - e * H;
  const int s = src[e], d = dst[e];
  const float alpha = p[t] / (dn[(size_t)d * H + h] + 1e-16f);
  const float* pl = xl + (size_t)s * (H * C) + h * C;
  float* po = acc + (size_t)d * (H * C) + h * C;
  if constexpr (C % 4 == 0) {
    const float4* pl4 = (const float4*)pl;
#pragma unroll
    for (int i = 0; i < C / 4; ++i) {
      const float4 a = pl4[i];
      atomicAdd(&po[4 * i + 0], alpha * a.x);
      atomicAdd(&po[4 * i + 1], alpha * a.y);
      atomicAdd(&po[4 * i + 2], alpha * a.z);
      atomicAdd(&po[4 * i + 3], alpha * a.w);
    }
  } else {
#pragma unroll
    for (int c = 0; c < C; ++c) atomicAdd(&po[c], alpha * pl[c]);
  }
}

// acc = leaky_relu(acc + bias, 0.01) in place  (layer-1 concat output + act)
__global__ void finalize_concat_act(float* __restrict__ acc,
                                    const float* __restrict__ bias,
                                    int total, int D) {
  const int t = blockIdx.x * blockDim.x + threadIdx.x;
  if (t >= total) return;
  float v = acc[t] + bias[t % D];
  acc[t] = (v > 0.0f) ? v : 0.01f * v;
}

// out[n][c] = mean_h(acc[n][h][c]) + bias[c]   (layer-2, concat=False)
template <int H, int C>
__global__ void finalize_mean(const float* __restrict__ acc,
                              const float* __restrict__ bias,
                              float* __restrict__ out, int total) {
  const int t = blockIdx.x * blockDim.x + threadIdx.x;
  if (t >= total) return;
  const int node = t / C;
  const int c = t - node * C;
  float s = 0.0f;
#pragma unroll
  for (int h = 0; h < H; ++h) s += acc[(size_t)node * (H * C) + h * C + c];
  out[t] = s * (1.0f / H) + bias[c];
}

__global__ void fill_f32(float* __restrict__ p, float v, int n) {
  const int t = blockIdx.x * blockDim.x + threadIdx.x;
  if (t < n) p[t] = v;
}

// ---------------------------------------------------------------------------
// Host-side launch
// ---------------------------------------------------------------------------
static inline int cdiv(int a, int b) { return (a + b - 1) / b; }

extern "C" void kernel_launch(void* const* d_in, const int* in_sizes, int n_in,
                              void* d_out, int out_size, void* d_ws, size_t ws_size,
                              hipStream_t stream) {
  const float* x    = (const float*)d_in[0];
  const int*   ei   = (const int*)d_in[1];
  const float* W1l  = (const float*)d_in[2];
  const float* b1l  = (const float*)d_in[3];
  const float* W1r  = (const float*)d_in[4];
  const float* b1r  = (const float*)d_in[5];
  const float* att1 = (const float*)d_in[6];
  const float* bias1= (const float*)d_in[7];
  const float* W2l  = (const float*)d_in[8];
  const float* b2l  = (const float*)d_in[9];
  const float* W2r  = (const float*)d_in[10];
  const float* b2r  = (const float*)d_in[11];
  const float* att2 = (const float*)d_in[12];
  const float* bias2= (const float*)d_in[13];
  float* out = (float*)d_out;

  const int N = in_sizes[0] / 9;   // 100000
  const int E = in_sizes[1] / 2;   // 1600000
  const int* src = ei;
  const int* dst = ei + E;

  constexpr int H = 4, C1 = 16, C2 = 9;
  constexpr int D1 = H * C1;       // 64
  constexpr int D2 = H * C2;       // 36

  // workspace layout (floats; every region is a multiple of 4 floats -> 16B
  // alignment preserved for float4 loads)
  float* w = (float*)d_ws;
  size_t o = 0;
  float* xl1  = w + o; o += (size_t)N * D1;
  float* xr1  = w + o; o += (size_t)N * D1;
  float* acc1 = w + o; o += (size_t)N * D1;   // becomes h after finalize
  float* xl2  = w + o; o += (size_t)N * D2;
  float* xr2  = w + o; o += (size_t)N * D2;
  float* acc2 = w + o; o += (size_t)N * D2;
  float* elog = w + o; o += (size_t)E * H;    // logits, then p
  float* mx   = w + o; o += (size_t)N * H;
  float* dn   = w + o; o += (size_t)N * H;
  (void)ws_size; (void)n_in; (void)out_size;

  const int BLK = 256;
  const int gemm_waves = cdiv(N, 16);
  const int gemm_grid  = cdiv(gemm_waves * 32, BLK);

  // ----- Layer 1 -----
  fill_f32<<<cdiv(N * H, BLK), BLK, 0, stream>>>(mx, -FLT_MAX, N * H);
  fill_f32<<<cdiv(N * H, BLK), BLK, 0, stream>>>(dn, 0.0f, N * H);
  fill_f32<<<cdiv(N * D1, BLK), BLK, 0, stream>>>(acc1, 0.0f, N * D1);

  gemm_bias_wmma<9, D1><<<gemm_grid, BLK, 0, stream>>>(x, W1l, b1l, xl1, N);
  gemm_bias_wmma<9, D1><<<gemm_grid, BLK, 0, stream>>>(x, W1r, b1r, xr1, N);

  edge_logits_max<H, C1><<<cdiv(E, BLK), BLK, 0, stream>>>(
      xl1, xr1, src, dst, att1, elog, mx, E);
  edge_exp_denom<H><<<cdiv(E * H, BLK), BLK, 0, stream>>>(elog, mx, dst, dn, E * H);
  edge_aggregate<H, C1><<<cdiv(E * H, BLK), BLK, 0, stream>>>(
      elog, dn, xl1, src, dst, acc1, E);
  finalize_concat_act<<<cdiv(N * D1, BLK), BLK, 0, stream>>>(acc1, bias1, N * D1, D1);

  // ----- Layer 2 -----
  fill_f32<<<cdiv(N * H, BLK), BLK, 0, stream>>>(mx, -FLT_MAX, N * H);
  fill_f32<<<cdiv(N * H, BLK), BLK, 0, stream>>>(dn, 0.0f, N * H);
  fill_f32<<<cdiv(N * D2, BLK), BLK, 0, stream>>>(acc2, 0.0f, N * D2);

  gemm_bias_wmma<D1, D2><<<gemm_grid, BLK, 0, stream>>>(acc1, W2l, b2l, xl2, N);
  gemm_bias_wmma<D1, D2><<<gemm_grid, BLK, 0, stream>>>(acc1, W2r, b2r, xr2, N);

  edge_logits_max<H, C2><<<cdiv(E, BLK), BLK, 0, stream>>>(
      xl2, xr2, src, dst, att2, elog, mx, E);
  edge_exp_denom<H><<<cdiv(E * H, BLK), BLK, 0, stream>>>(elog, mx, dst, dn, E * H);
  edge_aggregate<H, C2><<<cdiv(E * H, BLK), BLK, 0, stream>>>(
      elog, dn, xl2, src, dst, acc2, E);
  finalize_mean<H, C2><<<cdiv(N * C2, BLK), BLK, 0, stream>>>(acc2, bias2, out, N * C2);
}